// Graph2dConvolution_764504179074
// MI455X (gfx1250) — compile-verified
//
#include <hip/hip_runtime.h>
#include <math.h>

typedef __attribute__((ext_vector_type(2))) float v2f;
typedef __attribute__((ext_vector_type(4))) float v4f;
typedef __attribute__((ext_vector_type(8))) float v8f;

#define BATCH 8
#define CIN   128
#define COUT  128
#define HW    65536          // 256*256
#define KSEG  256            // BLOCK_NUM
#define BN_EPS 1e-5f

// LDS row stride for the staged x-tile: 64 cols + 8 pad so the two half-wave
// 16-wide read windows (rows k and k+2) hit disjoint banks (2*8 = 16 bank skew).
#define XS_STRIDE 72

// ---- workspace layout (float offsets into d_ws) ----
// accumulators (zeroed each launch):
#define OFF_SUMS  ((size_t)0)            // B*K*C = 262144
#define OFF_CNT   ((size_t)262144)       // B*K   = 2048
#define OFF_CS    ((size_t)264192)       // 128
#define OFF_CSS   ((size_t)264320)       // 128
#define N_ZERO    ((size_t)264448)
// derived buffers:
#define OFF_MEANS ((size_t)264448)       // 262144
#define OFF_Y     ((size_t)526592)       // 262144
#define OFF_SQ    ((size_t)788736)       // 2048
#define OFF_ADJW  ((size_t)790784)       // B*K*K = 524288
#define OFF_AGG   ((size_t)1315072)      // 262144
#define OFF_MU    ((size_t)1577216)      // 128
#define OFF_INV   ((size_t)1577344)      // 128

__device__ __forceinline__ v8f wmma_f32(v8f acc, v2f a, v2f b) {
  // D = A(16x4 f32) * B(4x16 f32) + C   -> v_wmma_f32_16x16x4_f32
  return __builtin_amdgcn_wmma_f32_16x16x4_f32(
      /*neg_a=*/false, a, /*neg_b=*/false, b,
      /*c_mod=*/(short)0, acc, /*reuse_a=*/false, /*reuse_b=*/false);
}

// ---------------------------------------------------------------- zero
__global__ __launch_bounds__(256) void k_zero(float* __restrict__ p, int n) {
  int i = blockIdx.x * 256 + threadIdx.x;
  if (i < n) p[i] = 0.0f;
}

// ---------------------------------------------------------------- stage 1: xo = W^T * X  (per batch)
// grid (HW/64, B), block 256 (8 waves).
// Workgroup computes a 128(d) x 64(p) output block: x-tile staged in LDS once,
// wave w covers d-rows [16w,16w+16) with 4 accumulators along pixels.
__global__ __launch_bounds__(256) void k_gemm_xo(const float* __restrict__ x,
                                                 const float* __restrict__ w,
                                                 float* __restrict__ xo) {
  __shared__ float xs[CIN * XS_STRIDE];       // 36 KB staged B-tile
  const int t    = threadIdx.x;
  const int b    = blockIdx.y;
  const int p0   = blockIdx.x * 64;
  const int wave = t >> 5;
  const int lane = t & 31;
  const int row  = lane & 15;
  const int hi   = lane >> 4;        // ISA fp32 A/B frag: lanes 16-31 hold K+2
  const int d0   = wave * 16;

  // ---- cooperative stage: x[b, 0..127, p0..p0+63] -> LDS (float4, coalesced)
  const float* xb = x + (size_t)b * CIN * HW + p0;
#pragma unroll
  for (int j = 0; j < 8; ++j) {
    const int i = t + 256 * j;       // float4 index: 2048 total
    const int r = i >> 4;            // k row
    const int c4 = (i & 15) << 2;    // col (floats)
    v4f v = *(const v4f*)(xb + (size_t)r * HW + c4);
    *(v4f*)(xs + r * XS_STRIDE + c4) = v;
  }
  __syncthreads();

  // ---- compute: 4 pixel sub-tiles per wave, A fragment reused 4x
  v8f acc0 = {0.f,0.f,0.f,0.f,0.f,0.f,0.f,0.f};
  v8f acc1 = acc0, acc2 = acc0, acc3 = acc0;
  for (int k = 0; k < CIN; k += 4) {
    const int ka = k + hi * 2;
    v2f a;
    a.x = w[(size_t)ka * COUT + d0 + row];        // A[m=d, k=c] = weight[c*COUT+d]
    a.y = w[(size_t)(ka + 1) * COUT + d0 + row];
    const float* bl0 = xs + ka * XS_STRIDE + row;
    const float* bl1 = xs + (ka + 1) * XS_STRIDE + row;
    v2f bf;
    bf.x = bl0[0];  bf.y = bl1[0];  acc0 = wmma_f32(acc0, a, bf);
    bf.x = bl0[16]; bf.y = bl1[16]; acc1 = wmma_f32(acc1, a, bf);
    bf.x = bl0[32]; bf.y = bl1[32]; acc2 = wmma_f32(acc2, a, bf);
    bf.x = bl0[48]; bf.y = bl1[48]; acc3 = wmma_f32(acc3, a, bf);
  }

  float* op = xo + (size_t)b * COUT * HW + p0;
#pragma unroll
  for (int r = 0; r < 8; ++r) {
    const size_t ro = (size_t)(d0 + r + hi * 8) * HW + row;
    op[ro]      = acc0[r];
    op[ro + 16] = acc1[r];
    op[ro + 32] = acc2[r];
    op[ro + 48] = acc3[r];
  }
}

// ---------------------------------------------------------------- segment counts
__global__ __launch_bounds__(256) void k_count(const int* __restrict__ idx,
                                               float* __restrict__ cnt) {
  int t = blockIdx.x * 256 + threadIdx.x;
  if (t >= BATCH * HW) return;
  int b = t / HW;
  atomicAdd(&cnt[b * KSEG + idx[t]], 1.0f);
}

// ---------------------------------------------------------------- segment sums (LDS accumulate)
// grid (16, B, 2), block 256; 64 channels per block (64 KB LDS), 4096 pixels per block
__global__ __launch_bounds__(256) void k_segsum(const float* __restrict__ xo,
                                                const int* __restrict__ idx,
                                                float* __restrict__ sums) {
  __shared__ float lsum[KSEG * 64];   // 64 KB
  const int t  = threadIdx.x;
  const int b  = blockIdx.y;
  const int c0 = blockIdx.z * 64;
  const int pb = blockIdx.x * 4096;
  for (int i = t; i < KSEG * 64; i += 256) lsum[i] = 0.0f;
  __syncthreads();

  for (int ch = 0; ch < 16; ++ch) {
    const int p  = pb + ch * 256 + t;
    const int sg = idx[(size_t)b * HW + p];
    const float* xp = xo + (size_t)b * COUT * HW + p;
#pragma unroll 4
    for (int c = 0; c < 64; ++c)
      atomicAdd(&lsum[sg * 64 + c], xp[(size_t)(c0 + c) * HW]);  // ds_add_f32
  }
  __syncthreads();
  for (int i = t; i < KSEG * 64; i += 256)
    atomicAdd(&sums[((size_t)b * KSEG + (i >> 6)) * COUT + c0 + (i & 63)], lsum[i]);
}

// ---------------------------------------------------------------- means = sums / max(cnt,1)
__global__ __launch_bounds__(256) void k_means(const float* __restrict__ sums,
                                               const float* __restrict__ cnt,
                                               float* __restrict__ means) {
  int t = blockIdx.x * 256 + threadIdx.x;
  if (t >= BATCH * KSEG * COUT) return;
  float c = cnt[t / COUT];
  means[t] = sums[t] / fmaxf(c, 1.0f);
}

// ---------------------------------------------------------------- y = means @ W_param  (M=2048,N=128,K=128)
__global__ __launch_bounds__(256) void k_gemm_y(const float* __restrict__ means,
                                                const float* __restrict__ Wp,
                                                float* __restrict__ y) {
  const int m0   = blockIdx.x * 16;
  const int wave = threadIdx.x >> 5;
  const int lane = threadIdx.x & 31;
  const int row = lane & 15, hi = lane >> 4;
  const int n0 = wave * 16;
  v8f acc = {0.f, 0.f, 0.f, 0.f, 0.f, 0.f, 0.f, 0.f};
  for (int k = 0; k < COUT; k += 4) {
    const int ka = k + hi * 2;
    v2f a, bf;
    a.x  = means[(size_t)(m0 + row) * COUT + ka];
    a.y  = means[(size_t)(m0 + row) * COUT + ka + 1];
    bf.x = Wp[(size_t)ka * COUT + n0 + row];
    bf.y = Wp[(size_t)(ka + 1) * COUT + n0 + row];
    acc = wmma_f32(acc, a, bf);
  }
#pragma unroll
  for (int r = 0; r < 8; ++r)
    y[(size_t)(m0 + r + hi * 8) * COUT + n0 + row] = acc[r];
}

// ---------------------------------------------------------------- sq[g] = |y_g|^2
__global__ __launch_bounds__(256) void k_sq(const float* __restrict__ y,
                                            float* __restrict__ sq) {
  int g = blockIdx.x * 256 + threadIdx.x;
  if (g >= BATCH * KSEG) return;
  const float* yr = y + (size_t)g * COUT;
  float s = 0.f;
#pragma unroll 8
  for (int c = 0; c < COUT; ++c) { float v = yr[c]; s += v * v; }
  sq[g] = s;
}

// ---------------------------------------------------------------- adjw = exp(-(|yi|^2+|yj|^2-2 Y Y^T)), diag 0
// grid (16, 2, B), block 256
__global__ __launch_bounds__(256) void k_adj(const float* __restrict__ y,
                                             const float* __restrict__ sq,
                                             float* __restrict__ adjw) {
  const int b    = blockIdx.z;
  const int i0   = blockIdx.x * 16;
  const int wave = threadIdx.x >> 5;
  const int j0   = (blockIdx.y * 8 + wave) * 16;
  const int lane = threadIdx.x & 31;
  const int row = lane & 15, hi = lane >> 4;
  const float* yb = y + (size_t)b * KSEG * COUT;
  v8f acc = {0.f, 0.f, 0.f, 0.f, 0.f, 0.f, 0.f, 0.f};
  for (int k = 0; k < COUT; k += 4) {
    const int ka = k + hi * 2;
    v2f a, bf;
    a.x  = yb[(size_t)(i0 + row) * COUT + ka];        // A[m=i,k]
    a.y  = yb[(size_t)(i0 + row) * COUT + ka + 1];
    bf.x = yb[(size_t)(j0 + row) * COUT + ka];        // B[k,n=j] = Y[j,k]
    bf.y = yb[(size_t)(j0 + row) * COUT + ka + 1];
    acc = wmma_f32(acc, a, bf);
  }
  const float* sqb = sq + (size_t)b * KSEG;
  float* ab = adjw + (size_t)b * KSEG * KSEG;
#pragma unroll
  for (int r = 0; r < 8; ++r) {
    const int i = i0 + r + hi * 8;
    const int j = j0 + row;
    const float q = sqb[i] + sqb[j] - 2.0f * acc[r];
    ab[(size_t)i * KSEG + j] = (i == j) ? 0.0f : expf(-q);
  }
}

// ---------------------------------------------------------------- agg = adjw @ means  (M=256,N=128,K=256 per b)
__global__ __launch_bounds__(256) void k_agg(const float* __restrict__ adjw,
                                             const float* __restrict__ means,
                                             float* __restrict__ agg) {
  const int b    = blockIdx.y;
  const int i0   = blockIdx.x * 16;
  const int wave = threadIdx.x >> 5;
  const int n0   = wave * 16;
  const int lane = threadIdx.x & 31;
  const int row = lane & 15, hi = lane >> 4;
  const float* ab = adjw + (size_t)b * KSEG * KSEG;
  const float* mb = means + (size_t)b * KSEG * COUT;
  v8f acc = {0.f, 0.f, 0.f, 0.f, 0.f, 0.f, 0.f, 0.f};
  for (int k = 0; k < KSEG; k += 4) {
    const int ka = k + hi * 2;
    v2f a, bf;
    a.x  = ab[(size_t)(i0 + row) * KSEG + ka];
    a.y  = ab[(size_t)(i0 + row) * KSEG + ka + 1];
    bf.x = mb[(size_t)ka * COUT + n0 + row];
    bf.y = mb[(size_t)(ka + 1) * COUT + n0 + row];
    acc = wmma_f32(acc, a, bf);
  }
  float* ob = agg + (size_t)b * KSEG * COUT;
#pragma unroll
  for (int r = 0; r < 8; ++r)
    ob[(size_t)(i0 + r + hi * 8) * COUT + n0 + row] = acc[r];
}

// ---------------------------------------------------------------- per-channel sum / sumsq of feat = xo + agg[gid]
// grid (B*COUT, 16), block 256
__global__ __launch_bounds__(256) void k_stats(const float* __restrict__ xo,
                                               const int* __restrict__ idx,
                                               const float* __restrict__ agg,
                                               float* __restrict__ cs,
                                               float* __restrict__ css) {
  const int bc = blockIdx.x;
  const int b = bc / COUT, c = bc % COUT;
  const int t = threadIdx.x;
  const int p0 = blockIdx.y * (HW / 16);
  const float* xoc = xo + (size_t)bc * HW;
  const int* ib = idx + (size_t)b * HW;
  const float* ag = agg + (size_t)b * KSEG * COUT + c;
  float s = 0.f, ss = 0.f;
  for (int p = p0 + t; p < p0 + HW / 16; p += 256) {
    __builtin_prefetch(xoc + p + 1024, 0, 1);   // global_prefetch_b8 ahead of stream
    const float f = xoc[p] + ag[(size_t)ib[p] * COUT];
    s += f; ss += f * f;
  }
  __shared__ float r1[256], r2[256];
  r1[t] = s; r2[t] = ss;
  __syncthreads();
  for (int o = 128; o > 0; o >>= 1) {
    if (t < o) { r1[t] += r1[t + o]; r2[t] += r2[t + o]; }
    __syncthreads();
  }
  if (t == 0) { atomicAdd(&cs[c], r1[0]); atomicAdd(&css[c], r2[0]); }
}

// ---------------------------------------------------------------- BN params
__global__ void k_bnparam(const float* __restrict__ cs, const float* __restrict__ css,
                          float* __restrict__ mu, float* __restrict__ inv) {
  int c = threadIdx.x;
  if (c >= COUT) return;
  const float n = (float)(BATCH * HW);
  const float m = cs[c] / n;
  const float v = css[c] / n - m * m;     // biased variance
  mu[c] = m;
  inv[c] = rsqrtf(v + BN_EPS);
}

// ---------------------------------------------------------------- final: out = gamma*(feat-mu)*inv + beta (in place over xo)
__global__ __launch_bounds__(256) void k_out(float* __restrict__ xo,
                                             const int* __restrict__ idx,
                                             const float* __restrict__ agg,
                                             const float* __restrict__ gamma,
                                             const float* __restrict__ beta,
                                             const float* __restrict__ mu,
                                             const float* __restrict__ inv) {
  const int bc = blockIdx.x;
  const int b = bc / COUT, c = bc % COUT;
  const int t = threadIdx.x;
  const int p0 = blockIdx.y * (HW / 16);
  float* xoc = xo + (size_t)bc * HW;
  const int* ib = idx + (size_t)b * HW;
  const float* ag = agg + (size_t)b * KSEG * COUT + c;
  const float g = gamma[c], be = beta[c], m = mu[c], iv = inv[c];
  for (int p = p0 + t; p < p0 + HW / 16; p += 256) {
    const float f = xoc[p] + ag[(size_t)ib[p] * COUT];
    xoc[p] = g * (f - m) * iv + be;
  }
}

// ================================================================ host
extern "C" void kernel_launch(void* const* d_in, const int* in_sizes, int n_in,
                              void* d_out, int out_size, void* d_ws, size_t ws_size,
                              hipStream_t stream) {
  const float* x      = (const float*)d_in[0];
  const int*   index  = (const int*)d_in[1];
  const float* weight = (const float*)d_in[2];
  const float* Wp     = (const float*)d_in[3];
  const float* gamma  = (const float*)d_in[4];
  const float* beta   = (const float*)d_in[5];

  float* xo = (float*)d_out;              // reuse output buffer as xo/feat scratch
  float* ws = (float*)d_ws;
  float* sums  = ws + OFF_SUMS;
  float* cnt   = ws + OFF_CNT;
  float* cs    = ws + OFF_CS;
  float* css   = ws + OFF_CSS;
  float* means = ws + OFF_MEANS;
  float* y     = ws + OFF_Y;
  float* sq    = ws + OFF_SQ;
  float* adjw  = ws + OFF_ADJW;
  float* agg   = ws + OFF_AGG;
  float* mu    = ws + OFF_MU;
  float* inv   = ws + OFF_INV;

  // zero accumulators
  k_zero<<<dim3((N_ZERO + 255) / 256), 256, 0, stream>>>(ws, (int)N_ZERO);
  // stage 1: channel-mix GEMM (WMMA fp32, LDS-staged x tile)
  k_gemm_xo<<<dim3(HW / 64, BATCH), 256, 0, stream>>>(x, weight, xo);
  // segment counts + sums + means
  k_count<<<dim3((BATCH * HW) / 256), 256, 0, stream>>>(index, cnt);
  k_segsum<<<dim3(16, BATCH, 2), 256, 0, stream>>>(xo, index, sums);
  k_means<<<dim3((BATCH * KSEG * COUT) / 256), 256, 0, stream>>>(sums, cnt, means);
  // affinity pipeline: y = means@Wp ; adj = exp(-||yi-yj||^2) ; agg = adj@means
  k_gemm_y<<<dim3((BATCH * KSEG) / 16), 256, 0, stream>>>(means, Wp, y);
  k_sq<<<dim3((BATCH * KSEG) / 256), 256, 0, stream>>>(y, sq);
  k_adj<<<dim3(KSEG / 16, 2, BATCH), 256, 0, stream>>>(y, sq, adjw);
  k_agg<<<dim3(KSEG / 16, BATCH), 256, 0, stream>>>(adjw, means, agg);
  // batch-norm: stats (recompute feat on the fly), params, normalize in place
  k_stats<<<dim3(BATCH * COUT, 16), 256, 0, stream>>>(xo, index, agg, cs, css);
  k_bnparam<<<1, 128, 0, stream>>>(cs, css, mu, inv);
  k_out<<<dim3(BATCH * COUT, 16), 256, 0, stream>>>(xo, index, agg, gamma, beta, mu, inv);
}